// CapsuleLayer_17480516895276
// MI455X (gfx1250) — compile-verified
//
#include <hip/hip_runtime.h>
#include <math.h>

typedef float v2f __attribute__((ext_vector_type(2)));
typedef float v8f __attribute__((ext_vector_type(8)));

#define B_   32
#define I_   32
#define C_   2048
#define U_   64
#define O_   32
#define CB   16                 // c-values per workgroup
#define NBLK (C_ / CB)          // 128 workgroups
#define XS_STRIDE 34            // padded i-stride (bank-conflict avoidance, 8B aligned)
#define XS_CSTR   (32 * XS_STRIDE)
#define S_ELEMS   (U_ * O_ * B_)   // 65536

__device__ __forceinline__ float wave_sum(float x) {
#pragma unroll
  for (int m = 16; m >= 1; m >>= 1) x += __shfl_xor(x, m, 32);
  return x;
}
__device__ __forceinline__ float wave_max(float x) {
#pragma unroll
  for (int m = 16; m >= 1; m >>= 1) x = fmaxf(x, __shfl_xor(x, m, 32));
  return x;
}

// MODE 1: coef = 1/64 (softmax of zero logits), no agree, no v.
// MODE 2: agree with v1, b_logits := agree (write), coef = softmax(agree).
// MODE 3: agree with v2, coef = softmax(b_logits + agree), no write.
template <int MODE>
__global__ void __launch_bounds__(1024)
caps_pass(const float* __restrict__ W, const float* __restrict__ X,
          const float* __restrict__ Vws, float* __restrict__ Blog,
          float* __restrict__ Part) {
  __shared__ float xs[8 * XS_CSTR];  // 8 staged c-columns of x: xs[cc][b][i] (padded)
  __shared__ float bl[U_];           // per-c routing logits row for softmax

  const int tid  = threadIdx.x;
  const int lane = tid & 31;
  const int w    = tid >> 5;     // wave 0..31; owns u = 2w, 2w+1
  const int l16  = lane & 15;
  const int lh   = lane >> 4;
  const int c0   = blockIdx.x * CB;

  // v fragments in D-matrix layout (modes 2,3): vf[j][nt][r]
  v8f vf[4][2];
  if (MODE >= 2) {
#pragma unroll
    for (int j = 0; j < 4; ++j) {
      const int m0 = w * 64 + j * 16;
#pragma unroll
      for (int nt = 0; nt < 2; ++nt) {
#pragma unroll
        for (int r = 0; r < 8; ++r)
          vf[j][nt][r] = Vws[(size_t)(m0 + r + 8 * lh) * B_ + nt * 16 + l16];
      }
    }
  }

  v8f sacc[4][2];
#pragma unroll
  for (int j = 0; j < 4; ++j)
#pragma unroll
    for (int nt = 0; nt < 2; ++nt)
#pragma unroll
      for (int r = 0; r < 8; ++r) sacc[j][nt][r] = 0.0f;

  for (int h = 0; h < 2; ++h) {
    __syncthreads();
    // stage x[:, :, c0+8h .. +8) into LDS; c is innermost in x -> coalesced
    for (int t = tid; t < 2048; t += 1024) {
      const int cc4 = (t & 1) * 4;
      const int bi  = t >> 1;  // b*32+i
      const float4 vx = *(const float4*)(X + (size_t)bi * C_ + c0 + 8 * h + cc4);
      float* dst = xs + (bi >> 5) * XS_STRIDE + (bi & 31) + (size_t)cc4 * XS_CSTR;
      dst[0 * XS_CSTR] = vx.x;
      dst[1 * XS_CSTR] = vx.y;
      dst[2 * XS_CSTR] = vx.z;
      dst[3 * XS_CSTR] = vx.w;
    }
    __syncthreads();

    for (int cc2 = 0; cc2 < 8; ++cc2) {
      const int c = c0 + 8 * h + cc2;

      // B fragments: B(k, n) = x[b=n, i=k, c]; lane<16 holds col n, (K0,K1); lane>=16 (K2,K3)
      v2f bf[2][8];
      const float* xcol = xs + (size_t)cc2 * XS_CSTR;
#pragma unroll
      for (int nt = 0; nt < 2; ++nt) {
        const int n = nt * 16 + l16;
#pragma unroll
        for (int kk = 0; kk < 8; ++kk) {
          const int k = kk * 4 + 2 * lh;
          bf[nt][kk] = *(const v2f*)(xcol + n * XS_STRIDE + k);
        }
      }

      // u_hat tiles via chained f32 WMMA (M=uo, N=b, K=i)
      v8f uh[4][2];
#pragma unroll
      for (int j = 0; j < 4; ++j) {
        const int m0 = w * 64 + j * 16;
        const float* wrow = W + ((size_t)c * 2048 + m0 + l16) * 32 + 2 * lh;
        v2f a[8];
#pragma unroll
        for (int kk = 0; kk < 8; ++kk) a[kk] = *(const v2f*)(wrow + kk * 4);
#pragma unroll
        for (int nt = 0; nt < 2; ++nt) {
          v8f acc;
#pragma unroll
          for (int r = 0; r < 8; ++r) acc[r] = 0.0f;
#pragma unroll
          for (int kk = 0; kk < 8; ++kk)
            acc = __builtin_amdgcn_wmma_f32_16x16x4_f32(
                false, a[kk], false, bf[nt][kk], (short)0, acc, false, false);
          uh[j][nt] = acc;
        }
      }

      float coef0, coef1;
      if (MODE == 1) {
        coef0 = coef1 = 1.0f / 64.0f;
      } else {
        // agreement: sum_{b,o} u_hat * v  (per u), then /B
        float p0 = 0.0f, p1 = 0.0f;
#pragma unroll
        for (int nt = 0; nt < 2; ++nt)
#pragma unroll
          for (int r = 0; r < 8; ++r) {
            p0 += uh[0][nt][r] * vf[0][nt][r] + uh[1][nt][r] * vf[1][nt][r];
            p1 += uh[2][nt][r] * vf[2][nt][r] + uh[3][nt][r] * vf[3][nt][r];
          }
        const float ag0 = wave_sum(p0) * (1.0f / (float)B_);
        const float ag1 = wave_sum(p1) * (1.0f / (float)B_);
        float bn0, bn1;
        if (MODE == 2) {
          bn0 = ag0;
          bn1 = ag1;
          if (lane == 0) {
            Blog[c * U_ + 2 * w]     = bn0;
            Blog[c * U_ + 2 * w + 1] = bn1;
          }
        } else {
          bn0 = Blog[c * U_ + 2 * w]     + ag0;
          bn1 = Blog[c * U_ + 2 * w + 1] + ag1;
        }
        if (lane == 0) {
          bl[2 * w]     = bn0;
          bl[2 * w + 1] = bn1;
        }
        __syncthreads();
        const float t0 = bl[lane];
        const float t1 = bl[lane + 32];
        const float mx = wave_max(fmaxf(t0, t1));
        const float sm = wave_sum(__expf(t0 - mx) + __expf(t1 - mx));
        __syncthreads();  // all reads of bl done before next c overwrites it
        coef0 = __expf(bn0 - mx) / sm;
        coef1 = __expf(bn1 - mx) / sm;
      }

      // s partial accumulation (wave-private registers, deterministic)
#pragma unroll
      for (int j = 0; j < 4; ++j) {
        const float cf = (j < 2) ? coef0 : coef1;
#pragma unroll
        for (int nt = 0; nt < 2; ++nt)
#pragma unroll
          for (int r = 0; r < 8; ++r) sacc[j][nt][r] += cf * uh[j][nt][r];
      }
    }
  }

  // write this block's s-partials in [uo][b] layout (lane-contiguous in b)
  float* pb = Part + (size_t)blockIdx.x * S_ELEMS;
#pragma unroll
  for (int j = 0; j < 4; ++j) {
    const int m0 = w * 64 + j * 16;
#pragma unroll
    for (int nt = 0; nt < 2; ++nt)
#pragma unroll
      for (int r = 0; r < 8; ++r)
        pb[(size_t)(m0 + r + 8 * lh) * B_ + nt * 16 + l16] = sacc[j][nt][r];
  }
}

// Reduce partials over blocks, squash over O; one wave per (b,u), lane = o.
__global__ void __launch_bounds__(1024)
caps_squash(const float* __restrict__ Part, float* __restrict__ Vws,
            float* __restrict__ Out, int writeOut) {
  const int tid  = threadIdx.x;
  const int lane = tid & 31;
  const int wid  = blockIdx.x * 32 + (tid >> 5);  // 0..2047
  const int b = wid >> 6;
  const int u = wid & 63;
  const int o = lane;
  const size_t idx = (size_t)(u * O_ + o) * B_ + b;
  float s = 0.0f;
  for (int blk = 0; blk < NBLK; ++blk) s += Part[(size_t)blk * S_ELEMS + idx];
  const float sq = wave_sum(s * s);
  const float v = s * sq / ((1.0f + sq) * sqrtf(sq + 1e-8f));
  Vws[idx] = v;                                      // [uo][b] layout for pass kernels
  if (writeOut) Out[((size_t)b * U_ + u) * O_ + o] = v;  // (B,U,O) output
}

extern "C" void kernel_launch(void* const* d_in, const int* in_sizes, int n_in,
                              void* d_out, int out_size, void* d_ws, size_t ws_size,
                              hipStream_t stream) {
  const float* x = (const float*)d_in[0];   // (B, I, C)
  const float* W = (const float*)d_in[1];   // (C, U, O, I)
  float* out = (float*)d_out;               // (B, U, O, 1)

  float* part = (float*)d_ws;                           // NBLK * 65536 floats (32 MB)
  float* vws  = part + (size_t)NBLK * S_ELEMS;          // 65536 floats, [uo][b]
  float* blog = vws + S_ELEMS;                          // C*U floats

  dim3 pg(NBLK), pb(1024);
  caps_pass<1><<<pg, pb, 0, stream>>>(W, x, nullptr, nullptr, part);  // s1
  caps_squash<<<64, 1024, 0, stream>>>(part, vws, out, 0);            // v1
  caps_pass<2><<<pg, pb, 0, stream>>>(W, x, vws, blog, part);         // agree1 + s2
  caps_squash<<<64, 1024, 0, stream>>>(part, vws, out, 0);            // v2
  caps_pass<3><<<pg, pb, 0, stream>>>(W, x, vws, blog, part);         // agree2 + s3
  caps_squash<<<64, 1024, 0, stream>>>(part, vws, out, 1);            // v3 -> d_out
}